// WaveInteractionAttention_73186242724045
// MI455X (gfx1250) — compile-verified
//
#include <hip/hip_runtime.h>
#include <math.h>
#include <stdint.h>

// Problem constants (match reference: B=1, S=512, HID=256, H=8, D=32)
#define S_LEN   512
#define HID     256
#define SCALE_F 0.17677669529663687f     // 1/sqrt(32)
#define LOG2E_F 1.4426950408889634f
#define INV2PI_F 0.15915494309189535f

typedef __attribute__((ext_vector_type(2))) float v2f;
typedef __attribute__((ext_vector_type(8))) float v8f;

// sigma(v) = 1/(1+e^-v) via v_exp_f32 + v_rcp_f32 (no IEEE divide sequence)
__device__ __forceinline__ float fast_sigmoid(float v) {
    return __builtin_amdgcn_rcpf(1.0f + __builtin_amdgcn_exp2f(-v * LOG2E_F));
}

// ---------------------------------------------------------------------------
// GEMM: out[M=512, N=256] = X[512,256] @ W[256,256]^T + bias   (nn.Linear)
// One wave32 computes one 16x16 tile with V_WMMA_F32_16X16X4_F32, K step 4.
//
// A (16x4 f32, ISA 7.12.2): lanes 0-15 -> M=lane, {K0,K1}; lanes 16-31 ->
//   M=lane-16, {K2,K3}  => per-lane float2 at X[m][k + half*2].
// B (4x16 f32): lanes 0-15 -> N=lane, {K0,K1}; lanes 16-31 -> N=lane-16,
//   {K2,K3}; B[k][n] = W[n][k], contiguous along k -> float2 loads.
// C/D (16x16 f32, 8 VGPRs): vgpr g -> M = g + 8*(lane>>4), N = lane&15.
//
// Epilogues store pre-normalized quantities for the fused attention kernel:
//  MODE 0 (q): omega_rev  = sigmoid(v) * PI / (2*PI) = 0.5*sigmoid(v)
//  MODE 1 (k): amp_pre    = sigmoid(v) * SCALE * log2(e)
//  MODE 2 (v): out0 = raw v ; out1 = phase_rev = 0.5*tanh(v) = sigmoid(2v)-0.5
//  MODE 3    : plain v (output projection)
// ---------------------------------------------------------------------------
template <int MODE>
__global__ __launch_bounds__(128)
void gemm_wmma_ep(const float* __restrict__ X, const float* __restrict__ W,
                  const float* __restrict__ bias,
                  float* __restrict__ out0, float* __restrict__ out1)
{
    const int wid  = blockIdx.x * 4 + (threadIdx.x >> 5);   // global wave id
    const int lane = threadIdx.x & 31;
    const int half = lane >> 4;        // 0: K pair {0,1}, 1: K pair {2,3}
    const int l15  = lane & 15;
    const int tilesN = HID / 16;       // 16 tiles in N
    const int row = (wid / tilesN) * 16;
    const int col = (wid % tilesN) * 16;

    const float* aptr = X + (size_t)(row + l15) * HID + half * 2;
    const float* bptr = W + (size_t)(col + l15) * HID + half * 2;

    v8f acc = {};
#pragma unroll 8
    for (int k = 0; k < HID; k += 4) {
        v2f a = *(const v2f*)(aptr + k);   // global_load_b64
        v2f b = *(const v2f*)(bptr + k);   // global_load_b64
        // 8 args: (neg_a, A, neg_b, B, c_mod, C, reuse_a, reuse_b)
        acc = __builtin_amdgcn_wmma_f32_16x16x4_f32(false, a, false, b,
                                                    (short)0, acc, false, false);
    }

    const int   n  = col + l15;
    const float bn = bias[n];
#pragma unroll
    for (int g = 0; g < 8; ++g) {
        const int m   = row + g + half * 8;
        const float v = acc[g] + bn;
        const size_t idx = (size_t)m * HID + n;
        if (MODE == 0) {
            out0[idx] = 0.5f * fast_sigmoid(v);                  // omega / 2pi
        } else if (MODE == 1) {
            out0[idx] = (SCALE_F * LOG2E_F) * fast_sigmoid(v);   // pre-scaled A
        } else if (MODE == 2) {
            out0[idx] = v;                                       // raw v
            out1[idx] = fast_sigmoid(2.0f * v) - 0.5f;           // phase / 2pi
        } else {
            out0[idx] = v;                                       // plain
        }
    }
}

// ---------------------------------------------------------------------------
// Fused wave-interference attention.
// One block per query position i (512 blocks), one thread per hidden column
// c (256 threads). Per key j:
//   logit2 = amp_pre[i,c] * cos(2pi*(omega_rev[i,c]*(i-j) + phase_rev[i,c]))
//   p = 2^logit2        (== exp(SCALE*A*cos(omega*(i-j)+phase)))
// |logit2| <= SCALE*log2(e) ~ 0.255 -> exp can never overflow, so one-pass
// softmax with no max subtraction. v rows are staged 32 at a time into LDS
// with GLOBAL_LOAD_ASYNC_TO_LDS_B128 (ASYNCcnt-tracked, no VGPR round trip).
// ---------------------------------------------------------------------------
#define VCHUNK 32

__global__ __launch_bounds__(HID)
void wave_attn(const float* __restrict__ omega_rev, const float* __restrict__ amp_pre,
               const float* __restrict__ phase_rev, const float* __restrict__ vraw,
               float* __restrict__ context)
{
    __shared__ __align__(16) float vtile[VCHUNK * HID];   // 32 KB of LDS

    const int i = blockIdx.x;
    const int c = threadIdx.x;
    const size_t ic = (size_t)i * HID + c;

    const float wr = omega_rev[ic];   // omega / 2pi   (revolutions per step)
    const float ap = amp_pre[ic];     // A * SCALE * log2(e)
    const float pr = phase_rev[ic];   // phase / 2pi

    const uint32_t lds_base = (uint32_t)(uintptr_t)(&vtile[0]);

    float den  = 0.0f;
    float num  = 0.0f;
    float posf = (float)i;            // (i - j), exact fp32 counter

    for (int j0 = 0; j0 < S_LEN; j0 += VCHUNK) {
        __syncthreads();              // previous chunk fully consumed
        // stage 32 rows (8192 floats = 2048 x b128) memory -> LDS, async
        {
            const uint64_t gbase = (uint64_t)(uintptr_t)(vraw + (size_t)j0 * HID);
#pragma unroll
            for (int kk = 0; kk < 8; ++kk) {
                const uint32_t idx4    = (uint32_t)threadIdx.x + (uint32_t)kk * HID;
                const uint32_t lds_off = lds_base + idx4 * 16u;
                const uint64_t gaddr   = gbase + (uint64_t)idx4 * 16u;
                asm volatile("global_load_async_to_lds_b128 %0, %1, off"
                             :: "v"(lds_off), "v"(gaddr) : "memory");
            }
            asm volatile("s_wait_asynccnt 0x0" ::: "memory");
        }
        __syncthreads();              // all waves' async writes visible

#pragma unroll 4
        for (int r = 0; r < VCHUNK; ++r) {
            const float ang = fmaf(wr, posf, pr);            // revolutions
            const float cn  = __builtin_amdgcn_cosf(ang);    // cos(2pi*ang)
            const float p   = __builtin_amdgcn_exp2f(ap * cn);
            den += p;
            num  = fmaf(p, vtile[r * HID + c], num);
            posf -= 1.0f;
        }
    }

    context[ic] = num * __builtin_amdgcn_rcpf(den);
}

// ---------------------------------------------------------------------------
// Host-side launcher.  d_ws layout (floats): omega | amp | phase | vraw | ctx
// = 5 * 512*256 * 4B = 2.62 MB.
// ---------------------------------------------------------------------------
extern "C" void kernel_launch(void* const* d_in, const int* in_sizes, int n_in,
                              void* d_out, int out_size, void* d_ws, size_t ws_size,
                              hipStream_t stream)
{
    (void)in_sizes; (void)n_in; (void)out_size; (void)ws_size;

    const float* x  = (const float*)d_in[0];
    const float* wq = (const float*)d_in[1];
    const float* bq = (const float*)d_in[2];
    const float* wk = (const float*)d_in[3];
    const float* bk = (const float*)d_in[4];
    const float* wv = (const float*)d_in[5];
    const float* bv = (const float*)d_in[6];
    const float* wo = (const float*)d_in[7];
    const float* bo = (const float*)d_in[8];
    float* out = (float*)d_out;

    float* ws = (float*)d_ws;
    const size_t N = (size_t)S_LEN * HID;
    float* omega = ws + 0 * N;
    float* amp   = ws + 1 * N;
    float* phase = ws + 2 * N;
    float* vraw  = ws + 3 * N;
    float* ctx   = ws + 4 * N;

    // (512/16)*(256/16) = 512 output tiles, 4 waves (128 thr) per block.
    const dim3 gemmGrid(128);
    const dim3 gemmBlock(128);

    hipLaunchKernelGGL((gemm_wmma_ep<0>), gemmGrid, gemmBlock, 0, stream,
                       x, wq, bq, omega, (float*)nullptr);
    hipLaunchKernelGGL((gemm_wmma_ep<1>), gemmGrid, gemmBlock, 0, stream,
                       x, wk, bk, amp, (float*)nullptr);
    hipLaunchKernelGGL((gemm_wmma_ep<2>), gemmGrid, gemmBlock, 0, stream,
                       x, wv, bv, vraw, phase);

    hipLaunchKernelGGL(wave_attn, dim3(S_LEN), dim3(HID), 0, stream,
                       omega, amp, phase, vraw, ctx);

    hipLaunchKernelGGL((gemm_wmma_ep<3>), gemmGrid, gemmBlock, 0, stream,
                       ctx, wo, bo, out, (float*)nullptr);
}